// CompressionAttention_28149215658107
// MI455X (gfx1250) — compile-verified
//
#include <hip/hip_runtime.h>
#include <hip/hip_bf16.h>
#include <math.h>
#include <stdint.h>

#define S_LEN   1024
#define B_SZ    2
#define DM      1024
#define N_HEADS 16
#define D_HEAD  64
#define C_LEN   32
#define M_ROWS  (S_LEN * B_SZ)          // 2048
#define PAIRS   (B_SZ * N_HEADS)        // 32

#define KC      32                      // K-chunk staged per double-buffer step
#define LDSP    36                      // padded row stride (floats): conflict-free, 16B-aligned rows

typedef float v2f __attribute__((ext_vector_type(2)));
typedef float v8f __attribute__((ext_vector_type(8)));

// ---------------------------------------------------------------------------
// CDNA5 async global->LDS copy (ASYNCcnt) + wait helpers.
// ---------------------------------------------------------------------------
__device__ __forceinline__ void async_b128(uint32_t lds_off, const float* g) {
  unsigned long long ga = (unsigned long long)(uintptr_t)g;
  asm volatile("global_load_async_to_lds_b128 %0, %1, off"
               :: "v"(lds_off), "v"(ga) : "memory");
}

template <int N>
__device__ __forceinline__ void wait_asynccnt() {
#if __has_builtin(__builtin_amdgcn_s_wait_asynccnt)
  __builtin_amdgcn_s_wait_asynccnt((unsigned short)N);
#else
  asm volatile("s_wait_asynccnt %0" :: "n"(N) : "memory");
#endif
}

// ---------------------------------------------------------------------------
// One 16x16 f32 tile of  A(16xK) * B(16xK)^T  via V_WMMA_F32_16X16X4_F32,
// operands streamed from registers (used by the small `down` GEMM).
// ---------------------------------------------------------------------------
__device__ __forceinline__ v8f wmma_nt_16x16(const float* __restrict__ a, int lda,
                                             const float* __restrict__ b, int ldb,
                                             int K, int lane) {
  const int r  = lane & 15;
  const int kh = (lane >> 4) << 1;
  const float* ap = a + (long)r * lda + kh;
  const float* bp = b + (long)r * ldb + kh;
  v8f acc = {0.f, 0.f, 0.f, 0.f, 0.f, 0.f, 0.f, 0.f};
  for (int k = 0; k < K; k += 4) {
    v2f af = *(const v2f*)(ap + k);
    v2f bf = *(const v2f*)(bp + k);
    acc = __builtin_amdgcn_wmma_f32_16x16x4_f32(false, af, false, bf,
                                                (short)0, acc, false, false);
  }
  return acc;
}

// ---------------------------------------------------------------------------
// C[M,N] = A[M,K] * W[N,K]^T + bias[N]   (f32 WMMA 16x16x4)
// Block = 8 waves, tile 128x64, requires M % 128 == 0, K % KC == 0.
// A/B tiles double-buffered in LDS via global_load_async_to_lds_b128;
// compute overlaps the next chunk's DMA (s_wait_asynccnt pipeline).
// ---------------------------------------------------------------------------
__global__ __launch_bounds__(256) void gemm_nt_bias_wmma(
    const float* __restrict__ A, const float* __restrict__ W,
    const float* __restrict__ bias, float* __restrict__ C,
    int M, int N, int K, int lda, int ldb, int ldc)
{
  __shared__ float As[2][128][LDSP];   // 2*128*36*4 = 36864 B
  __shared__ float Bs[2][64][LDSP];    // 2* 64*36*4 = 18432 B

  const int t    = threadIdx.x;
  const int lane = t & 31;
  const int wv   = t >> 5;
  const int m0   = blockIdx.x * 128;   // block row base
  const int n0   = blockIdx.y * 64;    // block col base
  (void)M; (void)N;

  const int r  = lane & 15;
  const int kh = (lane >> 4) << 1;
  const int wrow = wv * 16 + r;        // this lane's A row within the tile

  const uint32_t as0 = (uint32_t)(uintptr_t)&As[0][0][0];
  const uint32_t bs0 = (uint32_t)(uintptr_t)&Bs[0][0][0];

  const float* Abase = A + (long)m0 * lda;
  const float* Bbase = W + (long)n0 * ldb;

  // stage K-chunk k0 into LDS buffer `buf` (6 x b128 per thread: A=4, B=2)
  auto stage = [&](int k0, int buf) {
    const uint32_t abase = as0 + (uint32_t)buf * (128 * LDSP * 4);
    const uint32_t bbase = bs0 + (uint32_t)buf * ( 64 * LDSP * 4);
#pragma unroll
    for (int j = 0; j < 4; ++j) {                 // A: 128 rows x 8 segs
      const int seg = j * 256 + t;
      const int row = seg >> 3;
      const int c4  = (seg & 7) << 2;
      async_b128(abase + (uint32_t)(row * LDSP + c4) * 4,
                 Abase + (long)row * lda + k0 + c4);
    }
#pragma unroll
    for (int j = 0; j < 2; ++j) {                 // B: 64 rows x 8 segs
      const int seg = j * 256 + t;
      const int row = seg >> 3;
      const int c4  = (seg & 7) << 2;
      async_b128(bbase + (uint32_t)(row * LDSP + c4) * 4,
                 Bbase + (long)row * ldb + k0 + c4);
    }
  };

  v8f acc0 = {0.f,0.f,0.f,0.f,0.f,0.f,0.f,0.f};
  v8f acc1 = acc0, acc2 = acc0, acc3 = acc0;

  const int nch = K / KC;
  stage(0, 0);
  for (int c = 0; c < nch; ++c) {
    const int cur = c & 1;
    if (c + 1 < nch) {
      stage((c + 1) * KC, cur ^ 1);
      wait_asynccnt<6>();              // >=6 outstanding = next chunk; current done
    } else {
      wait_asynccnt<0>();
    }
    __syncthreads();                   // current chunk visible to all waves

#pragma unroll
    for (int kk = 0; kk < KC; kk += 4) {
      v2f af = *(const v2f*)&As[cur][wrow][kk + kh];
      v2f b0 = *(const v2f*)&Bs[cur][     r][kk + kh];
      v2f b1 = *(const v2f*)&Bs[cur][16 + r][kk + kh];
      v2f b2 = *(const v2f*)&Bs[cur][32 + r][kk + kh];
      v2f b3 = *(const v2f*)&Bs[cur][48 + r][kk + kh];
      acc0 = __builtin_amdgcn_wmma_f32_16x16x4_f32(false, af, false, b0, (short)0, acc0, false, false);
      acc1 = __builtin_amdgcn_wmma_f32_16x16x4_f32(false, af, false, b1, (short)0, acc1, false, false);
      acc2 = __builtin_amdgcn_wmma_f32_16x16x4_f32(false, af, false, b2, (short)0, acc2, false, false);
      acc3 = __builtin_amdgcn_wmma_f32_16x16x4_f32(false, af, false, b3, (short)0, acc3, false, false);
    }
    __syncthreads();                   // all waves done reading before overwrite
  }

  // C/D layout: VGPR rr -> row m0 + wv*16 + 8*(lane>=16) + rr, col n0 + 16*t + lane%16
  const int lm = (lane >> 4) << 3;
  const int ln = lane & 15;
#pragma unroll
  for (int tt = 0; tt < 4; ++tt) {
    v8f a = (tt == 0) ? acc0 : (tt == 1) ? acc1 : (tt == 2) ? acc2 : acc3;
    const int col = n0 + tt * 16 + ln;
    const float bb = bias ? bias[col] : 0.f;
#pragma unroll
    for (int rr = 0; rr < 8; ++rr) {
      const int row = m0 + wv * 16 + lm + rr;
      C[(long)row * ldc + col] = a[rr] + bb;
    }
  }
}

// ---------------------------------------------------------------------------
// down[pair][c][s] = scale * sum_d q_down[c][h*64+d] * k_d[s][b][h*64+d]
// One 16x16 WMMA tile per wave; grid = (PAIRS, 16), 8 waves/block.
// ---------------------------------------------------------------------------
__global__ __launch_bounds__(256) void down_gemm_wmma(
    const float* __restrict__ q_down, const float* __restrict__ k_d,
    float* __restrict__ wbuf, float alpha)
{
  const int pair = blockIdx.x;               // b*16 + h
  const int b = pair >> 4, h = pair & 15;
  const int lane = threadIdx.x & 31;
  const int wv   = threadIdx.x >> 5;
  const int tile = blockIdx.y * 8 + wv;      // 0..127: 2 m-tiles x 64 n-tiles
  const int m0 = (tile & 1) * 16;            // c dimension (C=32)
  const int n0 = (tile >> 1) * 16;           // s dimension (S=1024)

  const float* a  = q_down + (long)m0 * DM + h * D_HEAD;                 // lda=DM
  const float* bb = k_d + (long)n0 * (B_SZ * DM) + b * DM + h * D_HEAD;  // ldb=B*DM
  v8f acc = wmma_nt_16x16(a, DM, bb, B_SZ * DM, D_HEAD, lane);

  float* dr = wbuf + (long)pair * (C_LEN * S_LEN);
  const int lm = (lane >> 4) << 3;
  const int ln = lane & 15;
#pragma unroll
  for (int rr = 0; rr < 8; ++rr)
    dr[(long)(m0 + lm + rr) * S_LEN + (n0 + ln)] = alpha * acc[rr];
}

// ---------------------------------------------------------------------------
// Per (pair,c) row of `down` (length S): global max, w = exp(down - max)
// (in place), norm = inclusive cumsum(w).  One wave32 per row.
// ---------------------------------------------------------------------------
__global__ __launch_bounds__(32) void row_softmax_scan(
    float* __restrict__ w, float* __restrict__ norm)
{
  const long row = blockIdx.x;               // pair*C + c, 1024 rows
  float* dr = w    + row * S_LEN;
  float* nr = norm + row * S_LEN;
  const int lane = threadIdx.x;

  float m = -INFINITY;
  for (int s = lane; s < S_LEN; s += 32) m = fmaxf(m, dr[s]);
#pragma unroll
  for (int d = 16; d; d >>= 1) m = fmaxf(m, __shfl_xor(m, d));

  float run = 0.f;
  for (int s0 = 0; s0 < S_LEN; s0 += 32) {
    float v = expf(dr[s0 + lane] - m);
    float inc = v;
#pragma unroll
    for (int d = 1; d < 32; d <<= 1) {
      float t = __shfl_up(inc, d);
      if (lane >= d) inc += t;
    }
    float tot = __shfl(inc, 31);
    dr[s0 + lane] = v;
    nr[s0 + lane] = run + inc;
    run += tot;
  }
}

// ---------------------------------------------------------------------------
// Streaming compressed-KV scan: one block per (b,h).  Running 32x128 state
// in LDS; per position s: state += w[c][s]*[k'|v'](s), up_c = q.state_k/norm,
// softmax over c, out = sum_c p_c * state_v / norm.
// ---------------------------------------------------------------------------
__global__ __launch_bounds__(256) void mla_scan_attn(
    const float* __restrict__ q_u,  const float* __restrict__ vdk,
    const float* __restrict__ vdv,  const float* __restrict__ w,
    const float* __restrict__ norm, float* __restrict__ out)
{
  const int pair = blockIdx.x;               // b*16 + h
  const int b = pair >> 4, h = pair & 15;
  const int t = threadIdx.x;

  __shared__ float state[C_LEN * 128];       // [c][0:64)=K state, [64:128)=V state
  __shared__ float vrow[128];
  __shared__ float qrow[D_HEAD];
  __shared__ float wrow[C_LEN];
  __shared__ float nrow[C_LEN];
  __shared__ float prow[C_LEN];
  __shared__ float part[256];

  for (int i = t; i < C_LEN * 128; i += 256) state[i] = 0.f;
  const long base = (long)b * DM + h * D_HEAD;
  const float scale = 0.125f;                // 1/sqrt(64)
  const float* wr = w    + (long)pair * C_LEN * S_LEN;
  const float* nr = norm + (long)pair * C_LEN * S_LEN;
  __syncthreads();

  for (int s = 0; s < S_LEN; ++s) {
    const long off = (long)s * (B_SZ * DM) + base;
    if (t < 64)        qrow[t]        = q_u[off + t] * scale;
    else if (t < 128)  vrow[t - 64]   = vdk[off + (t - 64)];
    else if (t < 192)  vrow[t - 64]   = vdv[off + (t - 128)];
    else {
      const int c = t - 192;                 // 64 threads: 32 w + 32 norm
      if (c < C_LEN) wrow[c] = wr[(long)c * S_LEN + s];
      else           nrow[c - C_LEN] = nr[(long)(c - C_LEN) * S_LEN + s];
    }
    __syncthreads();

    // running state update: 4096 elements, 16 per thread
#pragma unroll
    for (int i = 0; i < 16; ++i) {
      const int idx = t + i * 256;
      state[idx] += wrow[idx >> 7] * vrow[idx & 127];
    }
    __syncthreads();

    // partial dots: thread t handles (c = t/8, e-range 8*(t%8)..+8)
    {
      const int c  = t >> 3;
      const int eg = (t & 7) << 3;
      const float* st = state + c * 128 + eg;
      float sum = 0.f;
#pragma unroll
      for (int e = 0; e < 8; ++e) sum += qrow[eg + e] * st[e];
      part[t] = sum;
    }
    __syncthreads();

    // wave 0: finish dots, divide by norm, softmax over c (32 lanes)
    if (t < C_LEN) {
      float u = 0.f;
#pragma unroll
      for (int j = 0; j < 8; ++j) u += part[t * 8 + j];
      u /= nrow[t];
      float m = u;
#pragma unroll
      for (int d = 16; d; d >>= 1) m = fmaxf(m, __shfl_xor(m, d));
      float p  = expf(u - m);
      float sm = p;
#pragma unroll
      for (int d = 16; d; d >>= 1) sm += __shfl_xor(sm, d);
      prow[t] = p / (sm * nrow[t]);          // fold 1/norm of v_u in here
    }
    __syncthreads();

    if (t < D_HEAD) {
      float o = 0.f;
#pragma unroll
      for (int c = 0; c < C_LEN; ++c) o += prow[c] * state[c * 128 + 64 + t];
      out[off + t] = o;
    }
    __syncthreads();                         // protect state/LDS for next s
  }
}

// ---------------------------------------------------------------------------
extern "C" void kernel_launch(void* const* d_in, const int* in_sizes, int n_in,
                              void* d_out, int out_size, void* d_ws, size_t ws_size,
                              hipStream_t stream) {
  (void)in_sizes; (void)n_in; (void)out_size; (void)ws_size;
  const float* x      = (const float*)d_in[0];
  const float* q_down = (const float*)d_in[1];
  const float* Wq = (const float*)d_in[2];
  const float* bq = (const float*)d_in[3];
  const float* Wk = (const float*)d_in[4];
  const float* bk = (const float*)d_in[5];
  const float* Wv = (const float*)d_in[6];
  const float* bv = (const float*)d_in[7];
  const float* Wo = (const float*)d_in[8];
  const float* bo = (const float*)d_in[9];
  float* outp = (float*)d_out;

  // workspace layout (floats)
  const long GE = (long)M_ROWS * DM;         // 2048*1024
  const long WE = (long)PAIRS * C_LEN * S_LEN;
  float* ws   = (float*)d_ws;
  float* q_u  = ws;
  float* k_d  = q_u  + GE;
  float* v_d  = k_d  + GE;
  float* v_dk = v_d  + GE;
  float* v_dv = v_dk + GE;
  float* wbuf = v_dv + GE;                   // down -> exp weights (in place)
  float* nbuf = wbuf + WE;
  float* attn = v_d;                         // reuse v_d after v_dk/v_dv done

  const dim3 gblk(M_ROWS / 128, DM / 64, 1); // (16,16)
  const dim3 blk(256);

  // x-projections (independent)
  gemm_nt_bias_wmma<<<gblk, blk, 0, stream>>>(x, Wq, bq, q_u, M_ROWS, DM, DM, DM, DM, DM);
  gemm_nt_bias_wmma<<<gblk, blk, 0, stream>>>(x, Wk, bk, k_d, M_ROWS, DM, DM, DM, DM, DM);
  gemm_nt_bias_wmma<<<gblk, blk, 0, stream>>>(x, Wv, bv, v_d, M_ROWS, DM, DM, DM, DM, DM);
  // second-level projections of v_d
  gemm_nt_bias_wmma<<<gblk, blk, 0, stream>>>(v_d, Wk, bk, v_dk, M_ROWS, DM, DM, DM, DM, DM);
  gemm_nt_bias_wmma<<<gblk, blk, 0, stream>>>(v_d, Wv, bv, v_dv, M_ROWS, DM, DM, DM, DM, DM);

  // down = scale * q_d . k_d^T  per (b,h)
  down_gemm_wmma<<<dim3(PAIRS, 16, 1), blk, 0, stream>>>(q_down, k_d, wbuf, 0.125f);
  // w = exp(down - rowmax), norm = cumsum(w)
  row_softmax_scan<<<dim3(PAIRS * C_LEN), dim3(32), 0, stream>>>(wbuf, nbuf);

  // streaming compressed-KV attention (writes attn = reused v_d buffer)
  mla_scan_attn<<<dim3(PAIRS), blk, 0, stream>>>(q_u, v_dk, v_dv, wbuf, nbuf, attn);

  // final projection: out = attn @ Wo^T + bo
  gemm_nt_bias_wmma<<<gblk, blk, 0, stream>>>(attn, Wo, bo, outp, M_ROWS, DM, DM, DM, DM, DM);
}